// DriverGeneGNN_83863531422321
// MI455X (gfx1250) — compile-verified
//
#include <hip/hip_runtime.h>
#include <math.h>

// ---------------------------------------------------------------------------
// DriverGeneGNN on MI455X (gfx1250, wave32).
// GEMMs use V_WMMA_F32_16X16X4_F32 (native f32 tensor op, no precision loss).
// Edge scatter (the bandwidth-bound part, ~5GB of HBM traffic) uses coalesced
// f32 global atomics, one thread per (edge, feature).
// BatchNorm is invariant to per-column constant shifts, so the conv biases
// (which feed directly into BN, and are zeros anyway) are dropped.
// ---------------------------------------------------------------------------

typedef float v2f __attribute__((ext_vector_type(2)));
typedef float v8f __attribute__((ext_vector_type(8)));

#define N_NODES 50000
#define N_EDGES 1600000
#define M_TILES 3125            // 50000 / 16 exactly

#define MODE_RELU 1
#define MODE_SAN  2
#define MODE_BN   4
#define MODE_BIAS 8

__device__ __forceinline__ float san_f(float x) {
    if (x != x) return 0.0f;                       // NaN -> 0
    if (x == __builtin_inff())  return 100.0f;     // +inf -> 100
    if (x == -__builtin_inff()) return -100.0f;    // -inf -> -100
    return x;
}

// ---------------- degree / dinv --------------------------------------------
__global__ void k_degree(const int* __restrict__ ei, float* __restrict__ deg, int E) {
    int stride = gridDim.x * blockDim.x;
    for (int e = blockIdx.x * blockDim.x + threadIdx.x; e < E; e += stride)
        atomicAdd(&deg[ei[E + e]], 1.0f);          // col = ei[1][e]
}

__global__ void k_dinv(float* __restrict__ deg, int n) {
    int stride = gridDim.x * blockDim.x;
    for (int i = blockIdx.x * blockDim.x + threadIdx.x; i < n; i += stride) {
        float d = deg[i];
        deg[i] = (d > 0.0f) ? rsqrtf(fmaxf(d, 1e-12f)) : 0.0f;
    }
}

// ---------------- edge scatter: agg[col] += dinv[row]*dinv[col]*m[row] -----
__global__ void k_scatter(const int* __restrict__ ei, const float* __restrict__ dinv,
                          const float* __restrict__ m, float* __restrict__ agg,
                          int E, int logF) {
    const int F = 1 << logF;
    long long total  = (long long)E << logF;
    long long stride = (long long)gridDim.x * blockDim.x;
    for (long long t = (long long)blockIdx.x * blockDim.x + threadIdx.x;
         t < total; t += stride) {
        int e = (int)(t >> logF);
        int f = (int)(t & (F - 1));
        int r = ei[e];
        int c = ei[E + e];
        float nrm = dinv[r] * dinv[c];
        atomicAdd(&agg[(size_t)c * F + f], nrm * m[(size_t)r * F + f]);
    }
}

// ---------------- per-column sum / sumsq for BatchNorm ---------------------
template <int F>
__global__ void k_colstats(const float* __restrict__ A, int rows, float* __restrict__ stats) {
    __shared__ float s_sum[256];
    __shared__ float s_sq[256];
    const int tid    = threadIdx.x;
    const int groups = 256 / F;          // 4 for F=64, 2 for F=128
    const int g = tid / F;
    const int f = tid % F;
    float s = 0.0f, s2 = 0.0f;
    for (int row = blockIdx.x * groups + g; row < rows; row += gridDim.x * groups) {
        float v = A[(size_t)row * F + f];
        s += v; s2 += v * v;
    }
    s_sum[tid] = s; s_sq[tid] = s2;
    __syncthreads();
    if (tid < F) {
        for (int gg = 1; gg < groups; ++gg) {
            s  += s_sum[gg * F + tid];
            s2 += s_sq[gg * F + tid];
        }
        atomicAdd(&stats[tid],     s);
        atomicAdd(&stats[F + tid], s2);
    }
}

__global__ void k_bnfinal(float* __restrict__ stats, int F, float invN) {
    int f = threadIdx.x;
    if (f < F) {
        float mu  = stats[f] * invN;
        float var = stats[F + f] * invN - mu * mu;   // biased variance
        stats[f]     = mu;
        stats[F + f] = rsqrtf(var + 1e-5f);
    }
}

// ---------------- layer-0 epilogue: h1 = san(relu(bn(agg)) + x) ------------
__global__ void k_post0(const float* __restrict__ agg, const float* __restrict__ x,
                        const float* __restrict__ g, const float* __restrict__ be,
                        const float* __restrict__ stats, float* __restrict__ h1, int n) {
    int stride = gridDim.x * blockDim.x;
    for (int i = blockIdx.x * blockDim.x + threadIdx.x; i < n; i += stride) {
        int f = i & 63;
        float v = agg[i];
        v = g[f] * (v - stats[f]) * stats[64 + f] + be[f];
        v = fmaxf(v, 0.0f) + x[i];
        h1[i] = san_f(v);
    }
}

// ---------------- WMMA f32 GEMM, one wave per 16x16 output tile ------------
// out[M x Nout] = A[M x K] @ B[K x Nout], optional fused epilogues.
__global__ void __launch_bounds__(256)
k_gemm_wmma(const float* __restrict__ A, int lda,
            const float* __restrict__ B, int ldb,
            const float* __restrict__ bias,
            const float* __restrict__ agg, int ldagg,
            const float* __restrict__ bng, const float* __restrict__ bnbe,
            const float* __restrict__ stats,       // mu[0..F), rsig[F..2F)
            float* __restrict__ out, int ldo,
            int mtiles, int ntiles, int K, int mode) {
    int wave = (blockIdx.x * blockDim.x + threadIdx.x) >> 5;
    if (wave >= mtiles * ntiles) return;           // wave-uniform exit, EXEC stays full
    int lane = threadIdx.x & 31;
    int tn = wave % ntiles;
    int tm = wave / ntiles;
    int nn = lane & 15;                            // M index for A, N index for B/C/D
    int kh = lane >> 4;                            // K half / row-half select

    // A 16x4 tile: lane (m=nn) holds K = k + 2*kh + {0,1}
    const float* Arow = A + (size_t)(tm * 16 + nn) * lda + 2 * kh;
    // B 4x16 tile: lane (n=nn) holds K = k + 2*kh + {0,1}
    const float* Bp = B + (size_t)(2 * kh) * ldb + tn * 16 + nn;

    v8f c = {0.f, 0.f, 0.f, 0.f, 0.f, 0.f, 0.f, 0.f};
    for (int k = 0; k < K; k += 4) {
        v2f a = *(const v2f*)(Arow + k);           // 8B-aligned pair load
        v2f b;
        b.x = Bp[(size_t)k * ldb];
        b.y = Bp[(size_t)(k + 1) * ldb];
        c = __builtin_amdgcn_wmma_f32_16x16x4_f32(
            /*neg_a=*/false, a, /*neg_b=*/false, b,
            /*c_mod=*/(short)0, c, /*reuse_a=*/false, /*reuse_b=*/false);
    }

    // C/D layout: VGPR r holds row r (lanes 0-15) / r+8 (lanes 16-31), col = lane&15
    int col     = tn * 16 + nn;
    int rowbase = tm * 16 + 8 * kh;
    int Nout    = ntiles * 16;
    float bcol = (mode & MODE_BIAS) ? bias[col] : 0.0f;
    float mu = 0.f, rs = 0.f, gg = 0.f, bb = 0.f;
    if (mode & MODE_BN) {
        mu = stats[col]; rs = stats[Nout + col]; gg = bng[col]; bb = bnbe[col];
    }
#pragma unroll
    for (int r = 0; r < 8; ++r) {
        int row = rowbase + r;
        float v = c[r] + bcol;
        if (mode & MODE_BN)
            v += gg * (agg[(size_t)row * ldagg + col] - mu) * rs + bb;
        if (mode & MODE_RELU) v = fmaxf(v, 0.0f);
        if (mode & MODE_SAN)  v = san_f(v);
        out[(size_t)row * ldo + col] = v;
    }
}

// ---------------- fc2 (128->2) + softmax, logits & probs to d_out ----------
__global__ void k_fc2_softmax(const float* __restrict__ h, const float* __restrict__ w,
                              const float* __restrict__ b, float* __restrict__ out, int n) {
    __shared__ float sw[256];                      // fc2_w is 128x2
    if (threadIdx.x < 256) sw[threadIdx.x] = w[threadIdx.x];
    __syncthreads();
    float b0 = b[0], b1 = b[1];
    int stride = gridDim.x * blockDim.x;
    for (int i = blockIdx.x * blockDim.x + threadIdx.x; i < n; i += stride) {
        const float* hr = h + (size_t)i * 128;
        float s0 = b0, s1 = b1;
#pragma unroll 4
        for (int j = 0; j < 128; ++j) {
            float hv = hr[j];
            s0 += hv * sw[2 * j];
            s1 += hv * sw[2 * j + 1];
        }
        out[2 * i]     = s0;
        out[2 * i + 1] = s1;
        float mx = fmaxf(s0, s1);
        float e0 = expf(s0 - mx), e1 = expf(s1 - mx);
        float inv = 1.0f / (e0 + e1);
        out[2 * n + 2 * i]     = e0 * inv;
        out[2 * n + 2 * i + 1] = e1 * inv;
    }
}

// ---------------------------------------------------------------------------
extern "C" void kernel_launch(void* const* d_in, const int* in_sizes, int n_in,
                              void* d_out, int out_size, void* d_ws, size_t ws_size,
                              hipStream_t stream) {
    (void)in_sizes; (void)n_in; (void)out_size; (void)ws_size;
    const float* x    = (const float*)d_in[0];
    const int*   eis[2] = { (const int*)d_in[1], (const int*)d_in[2] };
    const float* w0[2]   = { (const float*)d_in[3],  (const float*)d_in[8]  };
    const float* w1[2]   = { (const float*)d_in[5],  (const float*)d_in[10] };
    const float* res1[2] = { (const float*)d_in[7],  (const float*)d_in[12] };
    // b0/b1 (d_in[4,6,9,11]) feed directly into BatchNorm which cancels
    // per-column constant shifts (and they are zeros) -> not needed.
    const float* bng0 = (const float*)d_in[13];
    const float* bnb0 = (const float*)d_in[14];
    const float* bng1 = (const float*)d_in[15];
    const float* bnb1 = (const float*)d_in[16];
    const float* fc1w = (const float*)d_in[17];
    const float* fc1b = (const float*)d_in[18];
    const float* fc2w = (const float*)d_in[19];
    const float* fc2b = (const float*)d_in[20];

    // Workspace layout (1KB-aligned float sub-buffers), ~128 MB total.
    float* ws = (float*)d_ws;
    size_t off = 0;
    auto alloc = [&](size_t nfl) {
        float* p = ws + off;
        off += (nfl + 255) & ~(size_t)255;
        return p;
    };
    float* dinv  = alloc(N_NODES);                 // degree -> d^{-1/2}
    float* stats = alloc(256);                     // sum/sumsq -> mu/rsig
    float* bufA  = alloc((size_t)N_NODES * 64);    // m0, then h1
    float* bufC  = alloc((size_t)N_NODES * 64);    // agg0
    float* bufB  = alloc((size_t)N_NODES * 128);   // m1 (later reused as fc1 out)
    float* bufD  = alloc((size_t)N_NODES * 128);   // agg1
    float* hcat  = alloc((size_t)N_NODES * 256);   // [out0 | out1]
    float* hfc1  = bufB;                           // dead after both slices

    const int TB = 256;
    auto gemm = [&](const float* A, int lda, const float* B, int ldb,
                    const float* bias, const float* agg, int ldagg,
                    const float* bg, const float* bb, const float* st,
                    float* out, int ldo, int ntiles, int K, int mode) {
        int waves  = M_TILES * ntiles;
        int blocks = (waves + 7) / 8;              // 8 waves / 256-thread block
        k_gemm_wmma<<<blocks, TB, 0, stream>>>(A, lda, B, ldb, bias, agg, ldagg,
                                               bg, bb, st, out, ldo,
                                               M_TILES, ntiles, K, mode);
    };

    for (int s = 0; s < 2; ++s) {
        const int* ei = eis[s];

        // degree -> d^{-1/2}
        hipMemsetAsync(dinv, 0, N_NODES * sizeof(float), stream);
        k_degree<<<(N_EDGES + TB - 1) / TB, TB, 0, stream>>>(ei, dinv, N_EDGES);
        k_dinv<<<(N_NODES + TB - 1) / TB, TB, 0, stream>>>(dinv, N_NODES);

        // layer 0: m0 = x @ W0   (WMMA f32)
        gemm(x, 64, w0[s], 64, nullptr, nullptr, 0, nullptr, nullptr, nullptr,
             bufA, 64, 4, 64, 0);
        hipMemsetAsync(bufC, 0, (size_t)N_NODES * 64 * sizeof(float), stream);
        k_scatter<<<16384, TB, 0, stream>>>(ei, dinv, bufA, bufC, N_EDGES, 6);
        hipMemsetAsync(stats, 0, 2 * 64 * sizeof(float), stream);
        k_colstats<64><<<512, TB, 0, stream>>>(bufC, N_NODES, stats);
        k_bnfinal<<<1, 128, 0, stream>>>(stats, 64, 1.0f / N_NODES);
        k_post0<<<8192, TB, 0, stream>>>(bufC, x, bng0, bnb0, stats, bufA,
                                         N_NODES * 64);     // h1 in bufA

        // layer 1: m1 = h1 @ W1
        gemm(bufA, 64, w1[s], 128, nullptr, nullptr, 0, nullptr, nullptr, nullptr,
             bufB, 128, 8, 64, 0);
        hipMemsetAsync(bufD, 0, (size_t)N_NODES * 128 * sizeof(float), stream);
        k_scatter<<<16384, TB, 0, stream>>>(ei, dinv, bufB, bufD, N_EDGES, 7);
        hipMemsetAsync(stats, 0, 2 * 128 * sizeof(float), stream);
        k_colstats<128><<<512, TB, 0, stream>>>(bufD, N_NODES, stats);
        k_bnfinal<<<1, 128, 0, stream>>>(stats, 128, 1.0f / N_NODES);

        // h2 = san( bn(agg1) + h1 @ res1 )  -> hcat columns [s*128, s*128+128)
        gemm(bufA, 64, res1[s], 128, nullptr, bufD, 128, bng1, bnb1, stats,
             hcat + s * 128, 256, 8, 64, MODE_BN | MODE_SAN);
    }

    // fc1: relu(hcat @ fc1_w + fc1_b)
    gemm(hcat, 256, fc1w, 128, fc1b, nullptr, 0, nullptr, nullptr, nullptr,
         hfc1, 128, 8, 256, MODE_BIAS | MODE_RELU);

    // fc2 + softmax -> d_out = [logits (N*2) | probs (N*2)]
    k_fc2_softmax<<<512, TB, 0, stream>>>(hfc1, fc2w, fc2b, (float*)d_out, N_NODES);
}